// BaseCRF_86474871537673
// MI455X (gfx1250) — compile-verified
//
#include <hip/hip_runtime.h>

// CRF forward-backward for B=256, S=512, T=48 on gfx1250 (MI455X).
// Scan steps are reformulated as f32 GEMMs against exp(transitions) so the
// sequential recurrence runs on V_WMMA_F32_16X16X4_F32.
//
// d_ws layout: [ beta : S*B*T floats ][ z : B floats ]  (~25.2 MB)
// d_out: alpha written during forward scan, then overwritten in-place by
//        exp(alpha + beta - z) in the combine kernel (recomputed every call).

namespace {
constexpr int kB = 256;
constexpr int kS = 512;
constexpr int kT = 48;
constexpr int kTileB = 16;              // batches per wave
constexpr int kNT = 3;                  // 48 / 16 N-tiles
constexpr int kKC = 12;                 // 48 / 4  K-chunks
constexpr int kTStride = kT + 2;        // padded stride for exp(trans) in LDS
constexpr int kAStride = kT + 4;        // padded stride for A-operand staging
}

typedef float v2f __attribute__((ext_vector_type(2)));
typedef float v8f __attribute__((ext_vector_type(8)));

__global__ __launch_bounds__(32)
void crf_scan_kernel(const float* __restrict__ em,      // (B,S,T)
                     const int*   __restrict__ mask,    // (B,S)
                     const float* __restrict__ start_t, // (T)
                     const float* __restrict__ end_t,   // (T)
                     const float* __restrict__ trans,   // (T,T)
                     float* __restrict__ alpha_out,     // (S,B,T) -> d_out
                     float* __restrict__ beta_out,      // (S,B,T) -> ws
                     float* __restrict__ z_out)         // (B)     -> ws
{
  __shared__ float ldsT[kT * kTStride];      // exp(transitions), padded
  __shared__ float ldsA[kTileB * kAStride];  // per-step A-operand staging

  const int lane = threadIdx.x & 31;
  const int hlf  = lane >> 4;        // 0 / 1 (lane half)
  const int l    = lane & 15;
  const int b0   = blockIdx.x * kTileB;

  // exp(trans) into padded LDS (constant across all steps).
  for (int i = lane; i < kT * kT; i += 32) {
    const int r = i / kT, c = i - r * kT;
    ldsT[r * kTStride + c] = __expf(trans[i]);
  }
  __syncthreads();

  // State in WMMA C-layout: st[n][r] = state[batch r+8*hlf][tag 16*n+l]
  v8f st[kNT];

  if (blockIdx.y == 0) {
    // ============================ FORWARD ============================
    #pragma unroll
    for (int n = 0; n < kNT; ++n) {
      const int tag = 16 * n + l;
      const float s0 = start_t[tag];
      #pragma unroll
      for (int r = 0; r < 8; ++r) {
        const int bg = b0 + r + 8 * hlf;
        const float a0 = em[(size_t)bg * kS * kT + tag] + s0;
        st[n][r] = a0;
        alpha_out[(size_t)bg * kT + tag] = a0;   // s = 0 slice
      }
    }

    #pragma unroll 1
    for (int s = 1; s < kS; ++s) {
      // per-batch max over 48 tags (3 tiles + butterfly within 16-lane half)
      float mx[8];
      #pragma unroll
      for (int r = 0; r < 8; ++r)
        mx[r] = fmaxf(fmaxf(st[0][r], st[1][r]), st[2][r]);
      #pragma unroll
      for (int d = 1; d < 16; d <<= 1) {
        #pragma unroll
        for (int r = 0; r < 8; ++r)
          mx[r] = fmaxf(mx[r], __shfl_xor(mx[r], d, 32));
      }

      __syncthreads();
      #pragma unroll
      for (int n = 0; n < kNT; ++n)
        #pragma unroll
        for (int r = 0; r < 8; ++r)
          ldsA[(r + 8 * hlf) * kAStride + 16 * n + l] = __expf(st[n][r] - mx[r]);
      __syncthreads();

      // D[b,j] = sum_k expA[b,k] * expT[k,j]  via 12x3 WMMA f32 16x16x4
      v8f acc[kNT] = {};
      #pragma unroll
      for (int kk = 0; kk < kKC; ++kk) {
        const int k0 = 4 * kk + 2 * hlf;           // K pair for this lane half
        v2f a;
        a.x = ldsA[l * kAStride + k0];
        a.y = ldsA[l * kAStride + k0 + 1];
        #pragma unroll
        for (int n = 0; n < kNT; ++n) {
          v2f bf;
          bf.x = ldsT[k0 * kTStride + 16 * n + l];
          bf.y = ldsT[(k0 + 1) * kTStride + 16 * n + l];
          acc[n] = __builtin_amdgcn_wmma_f32_16x16x4_f32(
              false, a, false, bf, (short)0, acc[n], false, false);
        }
      }

      #pragma unroll
      for (int r = 0; r < 8; ++r) {
        const int bg = b0 + r + 8 * hlf;
        const float mk = (float)mask[(size_t)bg * kS + s];
        const size_t ebase = (size_t)bg * kS * kT + (size_t)s * kT;
        #pragma unroll
        for (int n = 0; n < kNT; ++n) {
          const int tag = 16 * n + l;
          const float e  = em[ebase + tag];
          const float nv = e + mx[r] + __logf(acc[n][r]);
          const float up = nv * mk + st[n][r] * (1.0f - mk);
          st[n][r] = up;
          alpha_out[(size_t)s * kB * kT + (size_t)bg * kT + tag] = up;
        }
      }
    }

    // z[b] = logsumexp_t( alpha[S-1][b][t] + end_t[t] )
    float et[kNT];
    #pragma unroll
    for (int n = 0; n < kNT; ++n) et[n] = end_t[16 * n + l];
    float mz[8], sz[8];
    #pragma unroll
    for (int r = 0; r < 8; ++r)
      mz[r] = fmaxf(fmaxf(st[0][r] + et[0], st[1][r] + et[1]), st[2][r] + et[2]);
    #pragma unroll
    for (int d = 1; d < 16; d <<= 1) {
      #pragma unroll
      for (int r = 0; r < 8; ++r)
        mz[r] = fmaxf(mz[r], __shfl_xor(mz[r], d, 32));
    }
    #pragma unroll
    for (int r = 0; r < 8; ++r)
      sz[r] = __expf(st[0][r] + et[0] - mz[r]) +
              __expf(st[1][r] + et[1] - mz[r]) +
              __expf(st[2][r] + et[2] - mz[r]);
    #pragma unroll
    for (int d = 1; d < 16; d <<= 1) {
      #pragma unroll
      for (int r = 0; r < 8; ++r)
        sz[r] += __shfl_xor(sz[r], d, 32);
    }
    if (l == 0) {
      #pragma unroll
      for (int r = 0; r < 8; ++r)
        z_out[b0 + r + 8 * hlf] = mz[r] + __logf(sz[r]);
    }
  } else {
    // ============================ BACKWARD ============================
    #pragma unroll
    for (int n = 0; n < kNT; ++n) {
      const int tag = 16 * n + l;
      const float e0 = end_t[tag];
      #pragma unroll
      for (int r = 0; r < 8; ++r) {
        const int bg = b0 + r + 8 * hlf;
        st[n][r] = e0;
        beta_out[(size_t)(kS - 1) * kB * kT + (size_t)bg * kT + tag] = e0;
      }
    }

    #pragma unroll 1
    for (int s = kS - 2; s >= 0; --s) {
      // u[b,k] = beta[b,k] + e[s+1][b,k];  m_b = max_k u
      float u[kNT][8];
      float mx[8];
      #pragma unroll
      for (int r = 0; r < 8; ++r) {
        const int bg = b0 + r + 8 * hlf;
        const size_t ebase = (size_t)bg * kS * kT + (size_t)(s + 1) * kT;
        #pragma unroll
        for (int n = 0; n < kNT; ++n)
          u[n][r] = st[n][r] + em[ebase + 16 * n + l];
        mx[r] = fmaxf(fmaxf(u[0][r], u[1][r]), u[2][r]);
      }
      #pragma unroll
      for (int d = 1; d < 16; d <<= 1) {
        #pragma unroll
        for (int r = 0; r < 8; ++r)
          mx[r] = fmaxf(mx[r], __shfl_xor(mx[r], d, 32));
      }

      __syncthreads();
      #pragma unroll
      for (int n = 0; n < kNT; ++n)
        #pragma unroll
        for (int r = 0; r < 8; ++r)
          ldsA[(r + 8 * hlf) * kAStride + 16 * n + l] = __expf(u[n][r] - mx[r]);
      __syncthreads();

      // D[b,j] = sum_k U[b,k] * expT[j,k]  (transposed B reads from LDS)
      v8f acc[kNT] = {};
      #pragma unroll
      for (int kk = 0; kk < kKC; ++kk) {
        const int k0 = 4 * kk + 2 * hlf;
        v2f a;
        a.x = ldsA[l * kAStride + k0];
        a.y = ldsA[l * kAStride + k0 + 1];
        #pragma unroll
        for (int n = 0; n < kNT; ++n) {
          v2f bf;
          bf.x = ldsT[(16 * n + l) * kTStride + k0];
          bf.y = ldsT[(16 * n + l) * kTStride + k0 + 1];
          acc[n] = __builtin_amdgcn_wmma_f32_16x16x4_f32(
              false, a, false, bf, (short)0, acc[n], false, false);
        }
      }

      #pragma unroll
      for (int r = 0; r < 8; ++r) {
        const int bg = b0 + r + 8 * hlf;
        const float mk = (float)mask[(size_t)bg * kS + s + 1];
        #pragma unroll
        for (int n = 0; n < kNT; ++n) {
          const float nv = mx[r] + __logf(acc[n][r]);
          const float up = nv * mk + st[n][r] * (1.0f - mk);
          st[n][r] = up;
          beta_out[(size_t)s * kB * kT + (size_t)bg * kT + 16 * n + l] = up;
        }
      }
    }
  }
}

__global__ __launch_bounds__(256)
void crf_combine_kernel(const float* __restrict__ beta,
                        const float* __restrict__ z,
                        float* __restrict__ out, int n)
{
  const int idx = blockIdx.x * blockDim.x + threadIdx.x;
  if (idx < n) {
    const int b = (idx / kT) % kB;
    out[idx] = __expf(out[idx] + beta[idx] - z[b]);
  }
}

extern "C" void kernel_launch(void* const* d_in, const int* in_sizes, int n_in,
                              void* d_out, int out_size, void* d_ws, size_t ws_size,
                              hipStream_t stream) {
  (void)in_sizes; (void)n_in; (void)out_size; (void)ws_size;
  const float* emissions = (const float*)d_in[0];
  const int*   mask      = (const int*)d_in[1];
  const float* start_t   = (const float*)d_in[2];
  const float* end_t     = (const float*)d_in[3];
  const float* trans     = (const float*)d_in[4];

  float* alpha = (float*)d_out;                       // (S,B,T)
  float* beta  = (float*)d_ws;                        // (S,B,T)
  float* z     = beta + (size_t)kS * kB * kT;         // (B)

  dim3 grid(kB / kTileB, 2);                          // y=0 fwd, y=1 bwd
  crf_scan_kernel<<<grid, 32, 0, stream>>>(emissions, mask, start_t, end_t,
                                           trans, alpha, beta, z);

  const int n = kS * kB * kT;
  crf_combine_kernel<<<(n + 255) / 256, 256, 0, stream>>>(beta, z, alpha, n);
}